// NodeClsHead_69982197121242
// MI455X (gfx1250) — compile-verified
//
#include <hip/hip_runtime.h>
#include <math.h>

typedef __attribute__((ext_vector_type(2))) float v2f;
typedef __attribute__((ext_vector_type(8))) float v8f;

#define DD 128
#define FF 384           // 3*D
#define CC 40            // NUM_CLS
#define FEPS 1e-7f
#define LDS_STRIDE 386   // 16 rows * 386 floats: avoids 64-bank conflicts on column reads

// ---------------- fused feature-build + WMMA GEMM: h = [xE | logH | logS] @ W ----------------
__global__ __launch_bounds__(32)
void feat_gemm_kernel(const float* __restrict__ xE,
                      const float* __restrict__ xH,
                      const float* __restrict__ xS,
                      const float* __restrict__ W,
                      float* __restrict__ h, int N) {
    __shared__ float xs[16 * LDS_STRIDE];
    __shared__ float sH[16];
    __shared__ float sS[16];

    const int lane  = threadIdx.x;      // wave32: one full wave per block
    const int node0 = blockIdx.x * 16;

    // per-row log-map scale factors
    if (lane < 16) {
        int node = min(node0 + lane, N - 1);
        float x0  = xH[(size_t)node * DD];
        float unh = sqrtf(fmaxf(x0 * x0 - 1.0f, FEPS));
        sH[lane]  = acoshf(fmaxf(x0, 1.0f)) / unh;
        float c   = xS[(size_t)node * DD];
        float uns = sqrtf(fmaxf(1.0f - c * c, FEPS));
        sS[lane]  = acosf(fminf(fmaxf(c, -1.0f), 1.0f)) / uns;
    }
    __syncthreads();

    // stage 16 x 384 feature tile into LDS
    for (int r = 0; r < 16; ++r) {
        int node = min(node0 + r, N - 1);
        const float* pe = xE + (size_t)node * DD;
        const float* ph = xH + (size_t)node * DD;
        const float* ps = xS + (size_t)node * DD;
        for (int j = lane; j < FF; j += 32) {   // each 32-chunk is branch-uniform
            float f;
            if (j < DD) {
                f = pe[j];
            } else if (j < 2 * DD) {
                int jj = j - DD;
                f = (jj == 0) ? 0.0f : sH[r] * ph[jj];
            } else {
                int jj = j - 2 * DD;
                f = (jj == 0) ? 0.0f : sS[r] * ps[jj];
            }
            xs[r * LDS_STRIDE + j] = f;
        }
    }
    __syncthreads();

    // WMMA f32 16x16x4: A = 16 rows of features, B = W tile, 3 N-tiles (40 cols -> 48 padded)
    const int half = lane >> 4;         // 0: K={0,1}, 1: K={2,3}
    const int l16  = lane & 15;
    const int col0 = l16;
    const int col1 = 16 + l16;
    const int col2 = 32 + l16;
    const bool ok2 = (col2 < CC);
    // branch-free padding for the third tile: clamp address, zero via multiply
    const int   c2 = ok2 ? col2 : col0;   // loop-invariant safe address
    const float m2 = ok2 ? 1.0f : 0.0f;   // loop-invariant mask

    v8f acc0 = {}, acc1 = {}, acc2 = {};
    for (int k = 0; k < FF; k += 4) {
        int kk = k + 2 * half;
        // A fragment: row M=l16, cols kk,kk+1 (8B aligned: kk even, stride even)
        v2f a = *(const v2f*)&xs[l16 * LDS_STRIDE + kk];
        const float* w0 = W + (size_t)kk * CC;       // B row kk
        const float* w1 = w0 + CC;                   // B row kk+1
        v2f b0, b1, b2;
        b0.x = w0[col0];      b0.y = w1[col0];
        b1.x = w0[col1];      b1.y = w1[col1];
        b2.x = w0[c2] * m2;   b2.y = w1[c2] * m2;
        acc0 = __builtin_amdgcn_wmma_f32_16x16x4_f32(false, a, false, b0, (short)0, acc0, false, false);
        acc1 = __builtin_amdgcn_wmma_f32_16x16x4_f32(false, a, false, b1, (short)0, acc1, false, false);
        acc2 = __builtin_amdgcn_wmma_f32_16x16x4_f32(false, a, false, b2, (short)0, acc2, false, false);
    }

    // D layout: VGPR r -> M = r + 8*half, lane -> N column
    #pragma unroll
    for (int r = 0; r < 8; ++r) {
        int node = node0 + r + 8 * half;
        if (node < N) {
            float* hp = h + (size_t)node * CC;
            hp[col0] = acc0[r];
            hp[col1] = acc1[r];
            if (ok2) hp[col2] = acc2[r];
        }
    }
}

// ---------------- degree / dinv / self-loop / edge scatter ----------------
__global__ void deg_init_kernel(float* __restrict__ deg, int N) {
    int i = blockIdx.x * blockDim.x + threadIdx.x;
    if (i < N) deg[i] = 1.0f;   // self-loop contributes 1
}

__global__ void deg_acc_kernel(const int* __restrict__ col, float* __restrict__ deg, int E) {
    int e = blockIdx.x * blockDim.x + threadIdx.x;
    if (e < E) unsafeAtomicAdd(&deg[col[e]], 1.0f);
}

__global__ void dinv_kernel(const float* __restrict__ deg, float* __restrict__ dinv, int N) {
    int i = blockIdx.x * blockDim.x + threadIdx.x;
    if (i < N) dinv[i] = rsqrtf(fmaxf(deg[i], 1.0f));   // deg >= 1 always
}

__global__ void self_kernel(const float* __restrict__ h, const float* __restrict__ dinv,
                            float* __restrict__ out, int total) {
    int idx = blockIdx.x * blockDim.x + threadIdx.x;
    if (idx < total) {
        float di = dinv[idx / CC];
        out[idx] = h[idx] * (di * di);   // self-loop term: norm = dinv[i]^2
    }
}

__global__ void scatter_kernel(const int* __restrict__ row, const int* __restrict__ col,
                               const float* __restrict__ h, const float* __restrict__ dinv,
                               float* __restrict__ out, int E) {
    int e = blockIdx.x * blockDim.x + threadIdx.x;
    if (e < E) {
        int r = row[e];
        int c = col[e];
        float nrm = dinv[r] * dinv[c];
        const float* hp = h + (size_t)r * CC;
        float* op = out + (size_t)c * CC;
        #pragma unroll
        for (int j = 0; j < CC; ++j) {
            unsafeAtomicAdd(&op[j], hp[j] * nrm);
        }
    }
}

extern "C" void kernel_launch(void* const* d_in, const int* in_sizes, int n_in,
                              void* d_out, int out_size, void* d_ws, size_t ws_size,
                              hipStream_t stream) {
    const float* xE = (const float*)d_in[0];
    const float* xH = (const float*)d_in[1];
    const float* xS = (const float*)d_in[2];
    const float* W  = (const float*)d_in[3];
    const int*   ei = (const int*)d_in[4];

    const int N = in_sizes[0] / DD;
    const int E = in_sizes[4] / 2;
    const int* row = ei;         // edge_index[0]
    const int* col = ei + E;     // edge_index[1]

    char*  ws   = (char*)d_ws;
    float* h    = (float*)ws;                                       // N*CC floats
    float* deg  = (float*)(ws + (size_t)N * CC * sizeof(float));    // N floats
    float* dinv = deg + N;                                          // N floats
    float* out  = (float*)d_out;

    const int nb16 = (N + 15) / 16;
    feat_gemm_kernel<<<nb16, 32, 0, stream>>>(xE, xH, xS, W, h, N);
    deg_init_kernel<<<(N + 255) / 256, 256, 0, stream>>>(deg, N);
    deg_acc_kernel<<<(E + 255) / 256, 256, 0, stream>>>(col, deg, E);
    dinv_kernel<<<(N + 255) / 256, 256, 0, stream>>>(deg, dinv, N);
    self_kernel<<<((N * CC) + 255) / 256, 256, 0, stream>>>(h, dinv, out, N * CC);
    scatter_kernel<<<(E + 255) / 256, 256, 0, stream>>>(row, col, h, dinv, out, E);
}